// GCNModel_24197845745728
// MI455X (gfx1250) — compile-verified
//
#include <hip/hip_runtime.h>
#include <hip/hip_bf16.h>

typedef __attribute__((ext_vector_type(2))) float v2f;
typedef __attribute__((ext_vector_type(8))) float v8f;

#define N_NODES   100000
#define N_EDGES   1600000
#define D_IN      128
#define HIDDEN    64
#define N_CLASSES 10
#define N_GRAPHS  512

// ---------------- utility ----------------
__global__ void zero_f32(float* __restrict__ p, int n) {
    int i = blockIdx.x * blockDim.x + threadIdx.x;
    if (i < n) p[i] = 0.0f;
}

__global__ void degree_kernel(const int* __restrict__ dst, float* __restrict__ deg, int n_edges) {
    int e = blockIdx.x * blockDim.x + threadIdx.x;
    if (e < n_edges) atomicAdd(&deg[dst[e]], 1.0f);
}

__global__ void rsqrt_kernel(float* __restrict__ deg_to_dinv, int n) {
    int i = blockIdx.x * blockDim.x + threadIdx.x;
    if (i < n) deg_to_dinv[i] = rsqrtf(deg_to_dinv[i] + 1.0f);  // +1 self-loop
}

// ---------------- fp32 WMMA GEMM: H[M,64] = X[M,K] @ W[K,64] ----------------
// Wave-per-16x16-tile, K stepped by 4 via V_WMMA_F32_16X16X4_F32.
// Block = 256 threads = 8 waves = 2 M-tiles x 4 N-tiles -> 32 rows per block.
// Requires M % 32 == 0 and K % 4 == 0 (holds: M=100000, K in {128,64}).
__global__ __launch_bounds__(256) void gemm_n64_wmma(const float* __restrict__ X,
                                                     const float* __restrict__ W,
                                                     float* __restrict__ H,
                                                     int M, int K) {
    const int lane  = threadIdx.x & 31;   // wave32
    const int wave  = threadIdx.x >> 5;   // 0..7
    const int half  = lane >> 4;          // 0: lanes 0-15, 1: lanes 16-31
    const int l16   = lane & 15;
    const int mtile = wave >> 2;          // 0..1
    const int ntile = wave & 3;           // 0..3
    const int m0  = blockIdx.x * 32 + mtile * 16;
    const int n0  = ntile * 16;
    const int row = m0 + l16;             // A-matrix row owned by this lane
    const int col = n0 + l16;             // B/C/D column owned by this lane

    v8f c = {};
    for (int kk = 0; kk < K; kk += 4) {
        const int ka = kk + 2 * half;     // this lane covers K = ka, ka+1
        // A 16x4 fragment: lane holds A[row][ka], A[row][ka+1] (8B-aligned)
        v2f a = *(const v2f*)(X + (size_t)row * K + ka);
        // B 4x16 fragment (mirrored layout): lane holds W[ka][col], W[ka+1][col]
        v2f b;
        b.x = W[(size_t)(ka + 0) * HIDDEN + col];
        b.y = W[(size_t)(ka + 1) * HIDDEN + col];
        c = __builtin_amdgcn_wmma_f32_16x16x4_f32(false, a, false, b,
                                                  (short)0, c, false, false);
    }
    // C/D 16x16: VGPR r, lane -> D[r + 8*half][l16]
#pragma unroll
    for (int r = 0; r < 8; ++r) {
        H[(size_t)(m0 + r + 8 * half) * HIDDEN + col] = c[r];
    }
}

// ---------------- aggregation ----------------
// agg[n][f] = h[n][f] * dinv[n]^2   (self-loop contribution seeds the buffer)
__global__ void init_agg(const float* __restrict__ h, const float* __restrict__ dinv,
                         float* __restrict__ agg, int total) {
    int i = blockIdx.x * blockDim.x + threadIdx.x;
    if (i < total) {
        float di = dinv[i >> 6];
        agg[i] = h[i] * di * di;
    }
}

// wave per edge; each lane handles 2 features (float2 gather, 2 f32 atomics).
// h[src] rows live in L2 (25.6MB << 192MB), so gather + GLOBAL_ATOMIC_ADD_F32
// both resolve in L2.
__global__ __launch_bounds__(256) void edge_scatter(const float* __restrict__ h,
                                                    const int* __restrict__ src,
                                                    const int* __restrict__ dst,
                                                    const float* __restrict__ dinv,
                                                    float* __restrict__ agg,
                                                    int n_edges) {
    int lane = threadIdx.x & 31;
    int e = blockIdx.x * 8 + (threadIdx.x >> 5);
    if (e >= n_edges) return;
    int s = src[e];
    int d = dst[e];
    float coef = dinv[s] * dinv[d];
    const float2 hv = *(const float2*)(h + (size_t)s * HIDDEN + lane * 2);
    float* out = agg + (size_t)d * HIDDEN + lane * 2;
    atomicAdd(out + 0, hv.x * coef);
    atomicAdd(out + 1, hv.y * coef);
}

__global__ void bias_relu(float* __restrict__ agg, const float* __restrict__ b, int total) {
    int i = blockIdx.x * blockDim.x + threadIdx.x;
    if (i < total) agg[i] = fmaxf(agg[i] + b[i & (HIDDEN - 1)], 0.0f);
}

// ---------------- pooling ----------------
__global__ __launch_bounds__(256) void pool_kernel(const float* __restrict__ h,
                                                   const int* __restrict__ batch,
                                                   float* __restrict__ sums,
                                                   float* __restrict__ cnt,
                                                   int n_nodes) {
    int lane = threadIdx.x & 31;
    int n = blockIdx.x * 8 + (threadIdx.x >> 5);
    if (n >= n_nodes) return;
    int g = batch[n];
    float2 v = *(const float2*)(h + (size_t)n * HIDDEN + lane * 2);
    atomicAdd(&sums[g * HIDDEN + lane * 2 + 0], v.x);
    atomicAdd(&sums[g * HIDDEN + lane * 2 + 1], v.y);
    if (lane == 0) atomicAdd(&cnt[g], 1.0f);
}

// ---------------- FC head + log_softmax ----------------
__global__ void head_kernel(const float* __restrict__ sums, const float* __restrict__ cnt,
                            const float* __restrict__ Wfc, const float* __restrict__ bfc,
                            float* __restrict__ out) {
    int g = blockIdx.x * blockDim.x + threadIdx.x;
    if (g >= N_GRAPHS) return;
    float inv = 1.0f / fmaxf(cnt[g], 1.0f);
    float logit[N_CLASSES];
#pragma unroll
    for (int c = 0; c < N_CLASSES; ++c) logit[c] = bfc[c];
    for (int k = 0; k < HIDDEN; ++k) {
        float p = sums[g * HIDDEN + k] * inv;
#pragma unroll
        for (int c = 0; c < N_CLASSES; ++c) logit[c] += p * Wfc[k * N_CLASSES + c];
    }
    float mx = logit[0];
#pragma unroll
    for (int c = 1; c < N_CLASSES; ++c) mx = fmaxf(mx, logit[c]);
    float se = 0.0f;
#pragma unroll
    for (int c = 0; c < N_CLASSES; ++c) se += __expf(logit[c] - mx);
    float lse = __logf(se) + mx;
#pragma unroll
    for (int c = 0; c < N_CLASSES; ++c) out[g * N_CLASSES + c] = logit[c] - lse;
}

extern "C" void kernel_launch(void* const* d_in, const int* in_sizes, int n_in,
                              void* d_out, int out_size, void* d_ws, size_t ws_size,
                              hipStream_t stream) {
    const float* x     = (const float*)d_in[0];
    const int*   eidx  = (const int*)d_in[1];
    const int*   batch = (const int*)d_in[2];
    const float* W1    = (const float*)d_in[3];
    const float* b1    = (const float*)d_in[4];
    const float* W2    = (const float*)d_in[5];
    const float* b2    = (const float*)d_in[6];
    const float* Wfc   = (const float*)d_in[7];
    const float* bfc   = (const float*)d_in[8];
    float* out = (float*)d_out;

    const int* src = eidx;              // edge_index[0]
    const int* dst = eidx + N_EDGES;    // edge_index[1]

    // workspace layout (floats, 256B-aligned regions)
    float* ws   = (float*)d_ws;
    float* dinv = ws;                                 // 100000 (also deg accum)
    float* bufA = ws + 102400;                        // 100000*64
    float* bufB = bufA + (size_t)N_NODES * HIDDEN;    // 100000*64
    float* pooled = bufB + (size_t)N_NODES * HIDDEN;  // 512*64
    float* cnt  = pooled + N_GRAPHS * HIDDEN;         // 512

    const int TOT = N_NODES * HIDDEN;

    // degrees -> dinv
    zero_f32<<<(N_NODES + 255) / 256, 256, 0, stream>>>(dinv, N_NODES);
    degree_kernel<<<(N_EDGES + 255) / 256, 256, 0, stream>>>(dst, dinv, N_EDGES);
    rsqrt_kernel<<<(N_NODES + 255) / 256, 256, 0, stream>>>(dinv, N_NODES);

    // layer 1: h1 = x @ W1  (WMMA fp32), then normalized aggregation + bias + relu
    gemm_n64_wmma<<<N_NODES / 32, 256, 0, stream>>>(x, W1, bufA, N_NODES, D_IN);
    init_agg<<<(TOT + 255) / 256, 256, 0, stream>>>(bufA, dinv, bufB, TOT);
    edge_scatter<<<N_EDGES / 8, 256, 0, stream>>>(bufA, src, dst, dinv, bufB, N_EDGES);
    bias_relu<<<(TOT + 255) / 256, 256, 0, stream>>>(bufB, b1, TOT);

    // layer 2: h2 = a1 @ W2 (WMMA fp32), aggregation + bias + relu
    gemm_n64_wmma<<<N_NODES / 32, 256, 0, stream>>>(bufB, W2, bufA, N_NODES, HIDDEN);
    init_agg<<<(TOT + 255) / 256, 256, 0, stream>>>(bufA, dinv, bufB, TOT);
    edge_scatter<<<N_EDGES / 8, 256, 0, stream>>>(bufA, src, dst, dinv, bufB, N_EDGES);
    bias_relu<<<(TOT + 255) / 256, 256, 0, stream>>>(bufB, b2, TOT);

    // global mean pool
    zero_f32<<<(N_GRAPHS * HIDDEN + 255) / 256, 256, 0, stream>>>(pooled, N_GRAPHS * HIDDEN);
    zero_f32<<<(N_GRAPHS + 255) / 256, 256, 0, stream>>>(cnt, N_GRAPHS);
    pool_kernel<<<(N_NODES + 7) / 8, 256, 0, stream>>>(bufB, batch, pooled, cnt, N_NODES);

    // FC + log_softmax
    head_kernel<<<(N_GRAPHS + 255) / 256, 256, 0, stream>>>(pooled, cnt, Wfc, bfc, out);
}